// CGLIRALayer_40939628265960
// MI455X (gfx1250) — compile-verified
//
#include <hip/hip_runtime.h>
#include <hip/hip_bf16.h>
#include <stdint.h>

// ---------------- problem constants (match reference) ----------------
#define NUM_USERS_C 40000
#define CG_BATCH    64
#define REG_LAMBDA_C 500.0f
#define MAX_ITER_C  30
#define TOL_C       1e-6f
#define GRAM_ROWS   128   // K-rows per wave in the WMMA gram kernel

typedef float v2f __attribute__((ext_vector_type(2)));
typedef float v8f __attribute__((ext_vector_type(8)));

// ---------------- small utility kernels ----------------
__global__ void k_fill_u32(uint32_t* p, uint32_t v, int n) {
  int i = blockIdx.x * 256 + threadIdx.x;
  if (i < n) p[i] = v;
}

__global__ void k_hist(const int* __restrict__ rows, const int* __restrict__ cols,
                       int nnz, int* rcnt, int* ccnt) {
  int i = blockIdx.x * 256 + threadIdx.x;
  if (i >= nnz) return;
  atomicAdd(&rcnt[rows[i]], 1);
  atomicAdd(&ccnt[cols[i]], 1);
}

// single-block exclusive scan (n up to ~64K); ptr has n+1 entries
__global__ void k_scan(const int* __restrict__ cnt, int* __restrict__ ptr, int n) {
  __shared__ int part[1024];
  int t = threadIdx.x;
  int chunk = (n + 1023) >> 10;
  int begin = t * chunk;
  int end   = begin + chunk; if (end > n) end = n; if (begin > n) begin = n;
  int s = 0;
  for (int i = begin; i < end; ++i) s += cnt[i];
  part[t] = s;
  __syncthreads();
  for (int off = 1; off < 1024; off <<= 1) {
    int v = (t >= off) ? part[t - off] : 0;
    __syncthreads();
    part[t] += v;
    __syncthreads();
  }
  int excl = part[t] - s;
  for (int i = begin; i < end; ++i) { ptr[i] = excl; excl += cnt[i]; }
  if (t == 1023) ptr[n] = part[1023];
}

__global__ void k_scatter(const int* __restrict__ rows, const int* __restrict__ cols,
                          const float* __restrict__ vals, int nnz,
                          const int* __restrict__ rptr, int* rfill,
                          int* csr_col, float* csr_val,
                          const int* __restrict__ cptr, int* cfill,
                          int* csc_row, float* csc_val) {
  int i = blockIdx.x * 256 + threadIdx.x;
  if (i >= nnz) return;
  int r = rows[i], c = cols[i];
  float v = vals[i];
  int p = rptr[r] + atomicAdd(&rfill[r], 1);
  csr_col[p] = c; csr_val[p] = v;
  int q = cptr[c] + atomicAdd(&cfill[c], 1);
  csc_row[q] = r; csc_val[q] = v;
}

// V[n*64+j] = Xb[j*NI + n]   (build x_t = X_batch^T)
__global__ void k_transpose_in(const float* __restrict__ Xb, float* __restrict__ V, int NI) {
  int i = blockIdx.x * 256 + threadIdx.x;
  if (i >= NI * CG_BATCH) return;
  int n = i >> 6, j = i & 63;
  V[i] = Xb[(size_t)j * NI + n];
}

// out[j*NI + n] = Xs[n*64+j]
__global__ void k_transpose_out(const float* __restrict__ Xs, float* __restrict__ out, int NI) {
  int i = blockIdx.x * 256 + threadIdx.x;
  if (i >= NI * CG_BATCH) return;
  int n = i >> 6, j = i & 63;
  out[(size_t)j * NI + n] = Xs[i];
}

// ---------------- SpMM kernels ----------------
// One wave per output row; lane handles 2 of the 64 batch columns.
// Index/value stream: cooperative coalesced 32-entry chunk loads + wave32
// __shfl broadcast (cuts idx/val L2 traffic 32x vs per-lane redundant loads),
// with global_prefetch of the next chunk overlapping the broadcast loop.
__global__ __launch_bounds__(128)
void k_spmm_gather(const int* __restrict__ ptr, const int* __restrict__ idx,
                   const float* __restrict__ val, const float* __restrict__ V,
                   float* __restrict__ out, int nrows) {
  int row = blockIdx.x * 4 + (threadIdx.x >> 5);
  if (row >= nrows) return;
  int lane = threadIdx.x & 31;
  float2 acc; acc.x = 0.f; acc.y = 0.f;
  int b = ptr[row], e = ptr[row + 1];
  for (int base = b; base < e; base += 32) {
    int k = base + lane;
    int cc = 0; float vv = 0.f;
    if (k < e) { cc = idx[k]; vv = val[k]; }
    if (base + 32 < e) {                       // overlap next index chunk
      __builtin_prefetch(idx + k + 32, 0, 1);  // -> global_prefetch_b8
      __builtin_prefetch(val + k + 32, 0, 1);
    }
    int cnt = e - base; if (cnt > 32) cnt = 32;
    for (int t = 0; t < cnt; ++t) {
      int   c = __shfl(cc, t, 32);
      float v = __shfl(vv, t, 32);
      float2 x = *(const float2*)(V + (size_t)c * 64 + lane * 2);
      acc.x = fmaf(v, x.x, acc.x);
      acc.y = fmaf(v, x.y, acc.y);
    }
  }
  *(float2*)(out + (size_t)row * 64 + lane * 2) = acc;
}

// out[col][j] = sum_k val[k]*tmp[ridx[k]][j] + lambda * Vadd[col][j]
__global__ __launch_bounds__(128)
void k_spmm_gather_lam(const int* __restrict__ ptr, const int* __restrict__ idx,
                       const float* __restrict__ val, const float* __restrict__ tmp,
                       const float* __restrict__ Vadd, float lam,
                       float* __restrict__ out, int ncols) {
  int col = blockIdx.x * 4 + (threadIdx.x >> 5);
  if (col >= ncols) return;
  int lane = threadIdx.x & 31;
  float2 acc; acc.x = 0.f; acc.y = 0.f;
  int b = ptr[col], e = ptr[col + 1];
  for (int base = b; base < e; base += 32) {
    int k = base + lane;
    int rr = 0; float vv = 0.f;
    if (k < e) { rr = idx[k]; vv = val[k]; }
    if (base + 32 < e) {
      __builtin_prefetch(idx + k + 32, 0, 1);
      __builtin_prefetch(val + k + 32, 0, 1);
    }
    int cnt = e - base; if (cnt > 32) cnt = 32;
    for (int t = 0; t < cnt; ++t) {
      int   r = __shfl(rr, t, 32);
      float v = __shfl(vv, t, 32);
      float2 x = *(const float2*)(tmp + (size_t)r * 64 + lane * 2);
      acc.x = fmaf(v, x.x, acc.x);
      acc.y = fmaf(v, x.y, acc.y);
    }
  }
  float2 pv = *(const float2*)(Vadd + (size_t)col * 64 + lane * 2);
  acc.x = fmaf(lam, pv.x, acc.x);
  acc.y = fmaf(lam, pv.y, acc.y);
  *(float2*)(out + (size_t)col * 64 + lane * 2) = acc;
}

// ---------------- WMMA gram-diagonal kernel ----------------
// diag[j] += sum_n A[n][j] * B[n][j], computed as diag of 16x16 tiles of A^T B
// via V_WMMA_F32_16X16X4_F32.  One wave per (K-chunk, 16-col tile).
__global__ __launch_bounds__(32)
void k_gram_diag(const float* __restrict__ A, const float* __restrict__ B,
                 float* __restrict__ diag, int nrows) {
  int lane    = threadIdx.x;          // 0..31, all active: EXEC must be all ones
  int tile    = blockIdx.y;           // 0..3 -> columns [16*tile, 16*tile+16)
  int colbase = tile * 16;
  int n0      = blockIdx.x * GRAM_ROWS;
  int nend    = n0 + GRAM_ROWS; if (nend > nrows) nend = nrows;
  int m  = lane & 15;                 // A: M index / B: N index
  int kb = (lane >> 4) * 2;           // K pair handled by this half-wave

#if __has_builtin(__builtin_amdgcn_wmma_f32_16x16x4_f32)
  v8f c = {};
#pragma unroll 2
  for (int n = n0; n + 3 < nend; n += 4) {
    v2f a, b;
    // A-matrix (16x4, M=rows of A^T): lane m, VGPRs hold K=kb,kb+1
    a.x = A[(size_t)(n + kb)     * 64 + colbase + m];
    a.y = A[(size_t)(n + kb + 1) * 64 + colbase + m];
    // B-matrix (4x16): VGPR v holds row K=v+kb, N striped across lanes
    b.x = B[(size_t)(n + kb)     * 64 + colbase + m];
    b.y = B[(size_t)(n + kb + 1) * 64 + colbase + m];
    c = __builtin_amdgcn_wmma_f32_16x16x4_f32(false, a, false, b,
                                              (short)0, c, false, false);
  }
  // extract diagonal of the 16x16 f32 C/D layout:
  //  m in [0,8):  VGPR m,   lane m        -> lanes 0..7,   c[lane]
  //  m in [8,16): VGPR m-8, lane m+16     -> lanes 24..31, c[lane-24]
  union { v8f v; float f[8]; } u; u.v = c;
  if (lane < 8)        atomicAdd(&diag[colbase + lane],            u.f[lane]);
  else if (lane >= 24) atomicAdd(&diag[colbase + (lane - 24) + 8], u.f[lane - 24]);
#else
  // VALU fallback (no WMMA available in this toolchain)
  if (lane < 16) {
    float s = 0.f;
    int col = colbase + lane;
    for (int n = n0; n < nend; ++n)
      s = fmaf(A[(size_t)n * 64 + col], B[(size_t)n * 64 + col], s);
    atomicAdd(&diag[col], s);
  }
#endif
}

// ---------------- CG scalar/state kernels ----------------
__global__ void k_scal_alpha(const float* __restrict__ Rs, const float* __restrict__ pAP,
                             const int* __restrict__ done, float* __restrict__ alpha) {
  int j = threadIdx.x;
  float a = Rs[j] / (pAP[j] + 1e-12f);
  alpha[j] = (*done) ? 0.f : a;   // alpha==0 freezes X and R exactly once converged
}

// float4-vectorized: X += a*P, R -= a*AP  (64 % 4 == 0 -> aligned column quads)
__global__ void k_update_xr(float4* __restrict__ X, float4* __restrict__ R,
                            const float4* __restrict__ P, const float4* __restrict__ AP,
                            const float* __restrict__ alpha, int n16) {
  int i = blockIdx.x * 256 + threadIdx.x;
  if (i >= n16) return;
  int c0 = (i * 4) & 63;
  float4 a = *(const float4*)(alpha + c0);
  float4 p = P[i], ap = AP[i], x = X[i], r = R[i];
  x.x = fmaf(a.x, p.x, x.x);  r.x = fmaf(-a.x, ap.x, r.x);
  x.y = fmaf(a.y, p.y, x.y);  r.y = fmaf(-a.y, ap.y, r.y);
  x.z = fmaf(a.z, p.z, x.z);  r.z = fmaf(-a.z, ap.z, r.z);
  x.w = fmaf(a.w, p.w, x.w);  r.w = fmaf(-a.w, ap.w, r.w);
  X[i] = x; R[i] = r;
}

__global__ void k_scal_beta(float* __restrict__ Rs_old, const float* __restrict__ Rs_new,
                            const int* __restrict__ done, float* __restrict__ beta,
                            int* __restrict__ conv) {
  __shared__ float mx[64];
  int j = threadIdx.x;
  int d = *done;
  float rn = Rs_new[j];
  beta[j] = rn / (Rs_old[j] + 1e-12f);
  if (!d) Rs_old[j] = rn;       // Rs frozen once done (matches jnp.where)
  mx[j] = rn;
  __syncthreads();
  for (int off = 32; off > 0; off >>= 1) {
    if (j < off) mx[j] = fmaxf(mx[j], mx[j + off]);
    __syncthreads();
  }
  if (j == 0) *conv = (!d && sqrtf(mx[0]) < TOL_C) ? 1 : 0;
}

__global__ void k_update_p(float4* __restrict__ P, const float4* __restrict__ R,
                           const float* __restrict__ beta, const int* __restrict__ done,
                           int n16) {
  if (*done) return;            // P frozen with pre-update done flag
  int i = blockIdx.x * 256 + threadIdx.x;
  if (i >= n16) return;
  int c0 = (i * 4) & 63;
  float4 bt = *(const float4*)(beta + c0);
  float4 p = P[i], r = R[i];
  p.x = fmaf(bt.x, p.x, r.x);
  p.y = fmaf(bt.y, p.y, r.y);
  p.z = fmaf(bt.z, p.z, r.z);
  p.w = fmaf(bt.w, p.w, r.w);
  P[i] = p;
}

__global__ void k_done_or(int* done, const int* conv) { if (*conv) *done = 1; }

// ---------------- host orchestration ----------------
extern "C" void kernel_launch(void* const* d_in, const int* in_sizes, int n_in,
                              void* d_out, int out_size, void* d_ws, size_t ws_size,
                              hipStream_t stream) {
  const float* Xb   = (const float*)d_in[0];
  const int*   rows = (const int*)d_in[1];
  const int*   cols = (const int*)d_in[2];
  const float* vals = (const float*)d_in[3];

  const int NI  = in_sizes[0] / CG_BATCH;   // 16384 items
  const int NNZ = in_sizes[1];              // 1,000,000
  const int NU  = NUM_USERS_C;              // 40000 users (reference constant)
  const int N64 = NI * CG_BATCH;
  const int N16 = N64 / 4;

  // carve workspace (256B aligned): ~44 MB total
  char* w = (char*)d_ws;
  auto carve = [&](size_t bytes) { void* p = (void*)w; w += (bytes + 255) & ~(size_t)255; return p; };
  float* P       = (float*)carve((size_t)N64 * 4);
  float* R       = (float*)carve((size_t)N64 * 4);
  float* Xs      = (float*)carve((size_t)N64 * 4);
  float* AP      = (float*)carve((size_t)N64 * 4);
  float* tmp     = (float*)carve((size_t)NU * 64 * 4);
  float* csr_val = (float*)carve((size_t)NNZ * 4);
  float* csc_val = (float*)carve((size_t)NNZ * 4);
  int*   csr_col = (int*)  carve((size_t)NNZ * 4);
  int*   csc_row = (int*)  carve((size_t)NNZ * 4);
  int*   row_ptr = (int*)  carve((size_t)(NU + 1) * 4);
  int*   col_ptr = (int*)  carve((size_t)(NI + 1) * 4);
  int*   row_aux = (int*)  carve((size_t)NU * 4);
  int*   col_aux = (int*)  carve((size_t)NI * 4);
  float* Rs_old  = (float*)carve(64 * 4);
  float* Rs_new  = (float*)carve(64 * 4);
  float* pAP     = (float*)carve(64 * 4);
  float* alpha   = (float*)carve(64 * 4);
  float* beta    = (float*)carve(64 * 4);
  int*   done    = (int*)  carve(64);
  int*   conv    = (int*)  carve(64);

  const int B256 = 256;
  dim3 gN64((N64 + B256 - 1) / B256);
  dim3 gN16((N16 + B256 - 1) / B256);
  dim3 gNNZ((NNZ + B256 - 1) / B256);
  dim3 gGram((NI + GRAM_ROWS - 1) / GRAM_ROWS, 4);

  // --- build CSR (by user row) and CSC (by item col) each call ---
  k_fill_u32<<<(NU + 255) / 256, B256, 0, stream>>>((uint32_t*)row_aux, 0u, NU);
  k_fill_u32<<<(NI + 255) / 256, B256, 0, stream>>>((uint32_t*)col_aux, 0u, NI);
  k_hist<<<gNNZ, B256, 0, stream>>>(rows, cols, NNZ, row_aux, col_aux);
  k_scan<<<1, 1024, 0, stream>>>(row_aux, row_ptr, NU);
  k_scan<<<1, 1024, 0, stream>>>(col_aux, col_ptr, NI);
  k_fill_u32<<<(NU + 255) / 256, B256, 0, stream>>>((uint32_t*)row_aux, 0u, NU);
  k_fill_u32<<<(NI + 255) / 256, B256, 0, stream>>>((uint32_t*)col_aux, 0u, NI);
  k_scatter<<<gNNZ, B256, 0, stream>>>(rows, cols, vals, NNZ,
                                       row_ptr, row_aux, csr_col, csr_val,
                                       col_ptr, col_aux, csc_row, csc_val);

  // --- RHS: y = S_mm(X_batch^T); use AP buffer to hold x_t ---
  k_transpose_in<<<gN64, B256, 0, stream>>>(Xb, AP, NI);
  k_spmm_gather<<<(NU + 3) / 4, 128, 0, stream>>>(row_ptr, csr_col, csr_val, AP, tmp, NU);
  k_spmm_gather_lam<<<(NI + 3) / 4, 128, 0, stream>>>(col_ptr, csc_row, csc_val, tmp,
                                                      AP, 0.0f, R, NI);
  // X0 = 0, P0 = R0 = y, Rs0 = diag(R^T R)
  hipMemcpyAsync(P, R, (size_t)N64 * 4, hipMemcpyDeviceToDevice, stream);
  k_fill_u32<<<gN64, B256, 0, stream>>>((uint32_t*)Xs, 0u, N64);
  k_fill_u32<<<1, 64, 0, stream>>>((uint32_t*)Rs_old, 0u, 64);
  k_fill_u32<<<1, 64, 0, stream>>>((uint32_t*)done, 0u, 16);
  k_fill_u32<<<1, 64, 0, stream>>>((uint32_t*)conv, 0u, 16);
  k_gram_diag<<<gGram, 32, 0, stream>>>(R, R, Rs_old, NI);

  // --- 30 CG iterations ---
  for (int it = 0; it < MAX_ITER_C; ++it) {
    k_spmm_gather<<<(NU + 3) / 4, 128, 0, stream>>>(row_ptr, csr_col, csr_val, P, tmp, NU);
    k_spmm_gather_lam<<<(NI + 3) / 4, 128, 0, stream>>>(col_ptr, csc_row, csc_val, tmp,
                                                        P, REG_LAMBDA_C, AP, NI);
    k_fill_u32<<<1, 64, 0, stream>>>((uint32_t*)pAP, 0u, 64);
    k_gram_diag<<<gGram, 32, 0, stream>>>(P, AP, pAP, NI);         // WMMA
    k_scal_alpha<<<1, 64, 0, stream>>>(Rs_old, pAP, done, alpha);
    k_update_xr<<<gN16, B256, 0, stream>>>((float4*)Xs, (float4*)R, (const float4*)P,
                                           (const float4*)AP, alpha, N16);
    k_fill_u32<<<1, 64, 0, stream>>>((uint32_t*)Rs_new, 0u, 64);
    k_gram_diag<<<gGram, 32, 0, stream>>>(R, R, Rs_new, NI);       // WMMA
    k_scal_beta<<<1, 64, 0, stream>>>(Rs_old, Rs_new, done, beta, conv);
    k_update_p<<<gN16, B256, 0, stream>>>((float4*)P, (const float4*)R, beta, done, N16);
    k_done_or<<<1, 1, 0, stream>>>(done, conv);
  }

  // --- output: Z = X^T as (batch, items) ---
  k_transpose_out<<<gN64, B256, 0, stream>>>(Xs, (float*)d_out, NI);
  (void)n_in; (void)out_size; (void)ws_size;
}